// Solver_60335700574499
// MI455X (gfx1250) — compile-verified
//
#include <hip/hip_runtime.h>
#include <stdint.h>

// ---------------- problem constants (match reference, fp32) ----------------
namespace {
constexpr int   kNX = 4096;
constexpr int   kNV = 1024;
constexpr float kLX = 2.0f;
constexpr float kLV = 12.0f;
constexpr float kDX = kLX / kNX;          // 4.8828125e-4
constexpr float kDV = kLV / kNV;          // 0.01171875
constexpr float kDT = 0.05f;
constexpr float kGamma = 0.29289321881345254f;  // 1 - 1/sqrt(2)
constexpr float kGDT = kGamma * kDT;
constexpr float kC2  = 2.0f * kGamma * kDT;     // 2*gamma*DT
}

__device__ __forceinline__ float xs_of(int i) {
  return fmaf((float)i, kDX, -0.5f * kLX + 0.5f * kDX);
}
__device__ __forceinline__ float vs_of(int j) {
  return fmaf((float)j, kDV, -0.5f * kLV + 0.5f * kDV);
}
__device__ __forceinline__ float h_of(float x) {
  const float ef = 0.0125f;   // midpt/2/20, midpt = LX/4 = 0.5
  return 1.0f / (1.0f + 0.5f * (expf((x - 0.5f) / ef) + expf((-x - 0.5f) / ef)));
}
__device__ __forceinline__ float minmodf(float a, float b) {
  return (a * b > 0.0f) ? ((fabsf(a) < fabsf(b)) ? a : b) : 0.0f;
}

// ---------------- K1: implicit LBO solve (Thomas) , one wave per x-row ------
// mode 0: rhs = r0        mode 1: rhs = r0 + DT*acc - 2*g*DT*ce
__global__ __launch_bounds__(32)
void k_thomas(const float* __restrict__ r0, const float* __restrict__ acc,
              const float* __restrict__ ce, float* __restrict__ out,
              float TA, float nu, int mode)
{
  __shared__ float sr[kNV];
  __shared__ float scp[kNV];
  const int row  = blockIdx.x;
  const int lane = threadIdx.x;
  const size_t base = (size_t)row * kNV;

  for (int k = lane; k < kNV; k += 32) {
    float r = r0[base + k];
    if (mode) r = fmaf(kDT, acc[base + k], r) - kC2 * ce[base + k];
    sr[k] = r;
  }
  __syncthreads();

  if (lane == 0) {
    const float h   = h_of(xs_of(row));
    const float th  = kGDT * h;
    const float dcc = nu * (0.5f * kDV - 2.0f * TA / kDV) / kDV;
    const float b   = 1.0f - th * dcc;
    float cp_prev = 0.0f, rp_prev = 0.0f;
    for (int j = 0; j < kNV; ++j) {
      float vlo = fmaf((float)j, kDV, -0.5f * kLV);        // v_{j-1/2}
      float dl = (j == 0)       ? 0.0f : nu * (TA / kDV - 0.5f * vlo) / kDV;
      float du = (j == kNV - 1) ? 0.0f : nu * (0.5f * (vlo + kDV) + TA / kDV) / kDV;
      float a = -th * dl;
      float c = -th * du;
      float denom = b - a * cp_prev;
      float inv = 1.0f / denom;
      float cp = c * inv;
      float rp = (sr[j] - a * rp_prev) * inv;
      scp[j] = cp; sr[j] = rp;
      cp_prev = cp; rp_prev = rp;
    }
    float xn = 0.0f;
    for (int j = kNV - 1; j >= 0; --j) {
      xn = sr[j] - scp[j] * xn;
      sr[j] = xn;
    }
  }
  __syncthreads();
  for (int k = lane; k < kNV; k += 32) out[base + k] = sr[k];
}

// ---------------- K2: charge density via WMMA ones-matmul reduction ---------
typedef float v2f __attribute__((ext_vector_type(2)));
typedef float v8f __attribute__((ext_vector_type(8)));

__global__ __launch_bounds__(32)
void k_moments(const float* __restrict__ fe, const float* __restrict__ fi,
               float* __restrict__ rho)
{
  const int tile = blockIdx.x;          // 16 x-rows per block, one wave
  const int lane = threadIdx.x;
  const int row  = tile * 16 + (lane & 15);
  const int koff = (lane >> 4) * 2;     // A-matrix 16x4 f32 lane layout
  const float* pe = fe + (size_t)row * kNV + koff;
  const float* pi = fi + (size_t)row * kNV + koff;
#if __has_builtin(__builtin_amdgcn_wmma_f32_16x16x4_f32)
  v2f ones = {1.0f, 1.0f};              // all-ones B => D[m][n] = row-sum(A)
  v8f se = {0.f, 0.f, 0.f, 0.f, 0.f, 0.f, 0.f, 0.f};
  v8f si = {0.f, 0.f, 0.f, 0.f, 0.f, 0.f, 0.f, 0.f};
  for (int kb = 0; kb < kNV; kb += 4) {
    v2f ae = *(const v2f*)(pe + kb);
    v2f ai = *(const v2f*)(pi + kb);
    se = __builtin_amdgcn_wmma_f32_16x16x4_f32(false, ae, false, ones, (short)0, se, false, false);
    si = __builtin_amdgcn_wmma_f32_16x16x4_f32(false, ai, false, ones, (short)0, si, false, false);
  }
  // C/D layout: VGPR j, lanes 0-15 -> M=j ; lanes 16-31 -> M=j+8
  if (lane == 0 || lane == 16) {
    const int rbase = tile * 16 + (lane >> 4) * 8;
    for (int j = 0; j < 8; ++j)
      rho[rbase + j] = kDV * (si[j] - se[j]);   // ZE=-1, ZI=+1
  }
#else
  if (lane < 16) {
    const float* qe = fe + (size_t)row * kNV;
    const float* qi = fi + (size_t)row * kNV;
    float s0 = 0.f, s1 = 0.f;
    for (int j = 0; j < kNV; ++j) { s0 += qe[j]; s1 += qi[j]; }
    rho[row] = kDV * (s1 - s0);
  }
#endif
}

// ---------------- K3: ion momentum flux at x=0 and x=NX-1 -------------------
__global__ __launch_bounds__(64)
void k_ionflux(const float* __restrict__ fi1, float* __restrict__ tot)
{
  __shared__ float red[64];
  const int t = threadIdx.x;
  const int lane = t & 31;
  const size_t base = (t >> 5) ? (size_t)(kNX - 1) * kNV : 0;
  float s = 0.f;
  for (int v = lane; v < kNV; v += 32) s = fmaf(fi1[base + v], vs_of(v), s);
  red[t] = s;
  __syncthreads();
  if (t == 0) {
    float s0 = 0.f, s1 = 0.f;
    for (int i = 0; i < 32; ++i) { s0 += red[i]; s1 += red[32 + i]; }
    tot[0] = (s1 - s0) * kDV;   // -flux[0] + flux[NX-1]
  }
}

// ---------------- K4: E = DX*cumsum(rho) - mean -----------------------------
__global__ __launch_bounds__(256)
void k_efield(const float* __restrict__ rho, float* __restrict__ E)
{
  __shared__ float part[256];
  const int t = threadIdx.x;
  float loc[16];
  float s = 0.f;
  for (int k = 0; k < 16; ++k) { loc[k] = rho[t * 16 + k]; s += loc[k]; }
  part[t] = s;
  __syncthreads();
  for (int off = 1; off < 256; off <<= 1) {     // inclusive scan of partials
    float mine = part[t];
    float add = (t >= off) ? part[t - off] : 0.f;
    __syncthreads();
    part[t] = mine + add;
    __syncthreads();
  }
  const float excl = part[t] - s;
  float run = excl, es = 0.f, ev[16];
  for (int k = 0; k < 16; ++k) { run += loc[k]; ev[k] = run * kDX; es += ev[k]; }
  __syncthreads();
  part[t] = es;
  __syncthreads();
  for (int off = 128; off > 0; off >>= 1) {
    if (t < off) part[t] += part[t + off];
    __syncthreads();
  }
  const float mean = part[0] * (1.0f / kNX);
  for (int k = 0; k < 16; ++k) E[t * 16 + k] = ev[k] - mean;
}

// ---------------- K5: fused Vlasov fluxes + explicit LBO + combine ----------
// Halo tile (16+4)x(64+4) loaded with CDNA5 async global->LDS copies.
__global__ __launch_bounds__(256)
void k_vlasov(const float* __restrict__ f1, const float* __restrict__ E,
              const float* __restrict__ tot, const float* __restrict__ f0,
              const float* __restrict__ fbase, const float* __restrict__ acc1,
              float* __restrict__ outA, float* __restrict__ outB,
              float TA, float nu, float aScale, int stage)
{
  __shared__ float tile[20 * 68];
  const int t  = threadIdx.x;
  const int v0 = blockIdx.x * 64;
  const int x0 = blockIdx.y * 16;

  for (int idx = t; idx < 20 * 68; idx += 256) {
    const int r  = idx / 68;
    const int c  = idx % 68;
    const int gx = (x0 - 2 + r) & (kNX - 1);    // periodic in x
    const int gv = v0 - 2 + c;                  // zero ghosts in v
    if ((unsigned)gv < (unsigned)kNV) {
      const uint64_t ga = (uint64_t)(uintptr_t)(f1 + (size_t)gx * kNV + gv);
      const uint32_t la = (uint32_t)(uintptr_t)(&tile[idx]);
      asm volatile("global_load_async_to_lds_b32 %0, %1, off"
                   :: "v"(la), "v"(ga) : "memory");
    } else {
      tile[idx] = 0.0f;
    }
  }
  asm volatile("s_wait_asynccnt 0x0" ::: "memory");
  __syncthreads();

  const float total = tot[0];
  const int cc  = (t & 63) + 2;              // local v (with halo)
  const int rr0 = (t >> 6) + 2;              // local x (with halo)
  const int v = v0 + (cc - 2);
  const float f0v = f0[(size_t)(kNX / 2) * kNV + v];
  const float cx  = vs_of(v);
  const float vlo = fmaf((float)v, kDV, -0.5f * kLV);
  const float dl  = (v == 0)       ? 0.f : nu * (TA / kDV - 0.5f * vlo) / kDV;
  const float du  = (v == kNV - 1) ? 0.f : nu * (0.5f * (vlo + kDV) + TA / kDV) / kDV;
  const float dcc = nu * (0.5f * kDV - 2.f * TA / kDV) / kDV;

#define TL(R, C) tile[(R) * 68 + (C)]
  for (int k = 0; k < 4; ++k) {
    const int r = rr0 + 4 * k;
    const int x = x0 + (r - 2);
    const float cv = aScale * E[x];          // OMEGA_C_TAU*Z/A*E
    const float X  = xs_of(x);
    const float h  = h_of(X);
    const float aX = fabsf(X);
    const float w  = (aX < 0.5f) ? (2.0f - 4.0f * aX) : 0.0f;

    const float fm2 = TL(r - 2, cc), fm1 = TL(r - 1, cc), fc = TL(r, cc);
    const float fp1 = TL(r + 1, cc), fp2 = TL(r + 2, cc);
    const float gm2 = TL(r, cc - 2), gm1 = TL(r, cc - 1);
    const float gp1 = TL(r, cc + 1), gp2 = TL(r, cc + 2);

    // x-direction MUSCL flux divergence, c = VS[v]
    const float sxm = minmodf(fc - fm1, fm1 - fm2);
    const float sx0 = minmodf(fp1 - fc, fc - fm1);
    const float sxp = minmodf(fp2 - fp1, fp1 - fc);
    const float flo = (cx > 0.f) ? (fm1 + 0.5f * sxm) : (fc - 0.5f * sx0);
    const float fhi = (cx > 0.f) ? (fc + 0.5f * sx0) : (fp1 - 0.5f * sxp);
    const float vdfdx = cx * (fhi - flo) / kDX;

    // v-direction MUSCL flux divergence, c = a(x)
    const float svm = minmodf(fc - gm1, gm1 - gm2);
    const float sv0 = minmodf(gp1 - fc, fc - gm1);
    const float svp = minmodf(gp2 - gp1, gp1 - fc);
    const float glo = (cv > 0.f) ? (gm1 + 0.5f * svm) : (fc - 0.5f * sv0);
    const float ghi = (cv > 0.f) ? (fc + 0.5f * sv0) : (gp1 - 0.5f * svp);
    const float edfdv = cv * (ghi - glo) / kDV;

    const float ce = h * (dcc * fc + dl * gm1 + du * gp1);   // coll_expl
    const float ne = -vdfdx - edfdv + total * w * f0v;       // vlasov rhs

    const size_t o = (size_t)x * kNV + v;
    if (stage == 1) { outA[o] = ne + ce; outB[o] = ce; }
    else            { outA[o] = fbase[o] + 0.5f * kDT * (acc1[o] + ne + ce); }
  }
#undef TL
}

// ---------------- host-side pipeline ----------------------------------------
extern "C" void kernel_launch(void* const* d_in, const int* in_sizes, int n_in,
                              void* d_out, int out_size, void* d_ws, size_t ws_size,
                              hipStream_t stream)
{
  (void)in_sizes; (void)n_in; (void)out_size; (void)ws_size;
  const float* fe  = (const float*)d_in[0];
  const float* fi  = (const float*)d_in[1];
  const float* f0e = (const float*)d_in[2];
  const float* f0i = (const float*)d_in[3];
  const size_t S = (size_t)kNX * kNV;
  float* out_e = (float*)d_out;
  float* out_i = out_e + S;

  float* ws  = (float*)d_ws;
  float* fe1 = ws;            // also holds fe2 in stage 2
  float* fi1 = ws + S;
  float* Ce  = ws + 2 * S;
  float* Ci  = ws + 3 * S;
  float* Ae  = ws + 4 * S;    // Acc = N + C
  float* Ai  = ws + 5 * S;
  float* rho = ws + 6 * S;
  float* Ef  = rho + kNX;
  float* tot = Ef + kNX;

  const float TAe = 1.0f,  nue = 1.0f, aSe = -1.0f;   // ZE/AE
  const float TAi = 0.01f, nui = 0.1f, aSi = 0.01f;   // ZI/AI

  dim3 gT(kNX), bT(32);
  dim3 gM(kNX / 16), bM(32);
  dim3 gV(kNV / 64, kNX / 16), bV(256);

  // ---- stage 1 ----
  k_thomas<<<gT, bT, 0, stream>>>(fe, nullptr, nullptr, fe1, TAe, nue, 0);
  k_thomas<<<gT, bT, 0, stream>>>(fi, nullptr, nullptr, fi1, TAi, nui, 0);
  k_moments<<<gM, bM, 0, stream>>>(fe1, fi1, rho);
  k_ionflux<<<1, 64, 0, stream>>>(fi1, tot);
  k_efield<<<1, 256, 0, stream>>>(rho, Ef);
  k_vlasov<<<gV, bV, 0, stream>>>(fe1, Ef, tot, f0e, nullptr, nullptr, Ae, Ce, TAe, nue, aSe, 1);
  k_vlasov<<<gV, bV, 0, stream>>>(fi1, Ef, tot, f0i, nullptr, nullptr, Ai, Ci, TAi, nui, aSi, 1);
  // ---- stage 2 (rhs composed on the fly: f + DT*Acc1 - 2gDT*C1) ----
  k_thomas<<<gT, bT, 0, stream>>>(fe, Ae, Ce, fe1, TAe, nue, 1);
  k_thomas<<<gT, bT, 0, stream>>>(fi, Ai, Ci, fi1, TAi, nui, 1);
  k_moments<<<gM, bM, 0, stream>>>(fe1, fi1, rho);
  k_ionflux<<<1, 64, 0, stream>>>(fi1, tot);
  k_efield<<<1, 256, 0, stream>>>(rho, Ef);
  k_vlasov<<<gV, bV, 0, stream>>>(fe1, Ef, tot, f0e, fe, Ae, out_e, nullptr, TAe, nue, aSe, 2);
  k_vlasov<<<gV, bV, 0, stream>>>(fi1, Ef, tot, f0i, fi, Ai, out_i, nullptr, TAi, nui, aSi, 2);
}